// BiLstmCrfNetwork_88407606821376
// MI455X (gfx1250) — compile-verified
//
#include <hip/hip_runtime.h>
#include <hip/hip_bf16.h>

#define SEQ   16384
#define EMB   50
#define EMBP  52            // EMB padded to a multiple of 4 (K-dim for WMMA)
#define HID   300
#define G4    1200          // 4*HID, torch gate order i,f,g,o
#define H2    600           // 2*HID
#define TT    20
#define TTP   32            // tagset padded to 2 WMMA column tiles
#define START_TAG 18
#define STOP_TAG  19
#define NEGV  (-10000.0f)

typedef __attribute__((ext_vector_type(2)))  float v2f;
typedef __attribute__((ext_vector_type(8)))  float v8f;

__device__ __forceinline__ float sigmoidf_(float x) { return 1.0f / (1.0f + __expf(-x)); }

// ---------------------------------------------------------------------------
// K0: embedding gather with zero-padded K: emb_p[t*52+k] (k>=50 -> 0)
// ---------------------------------------------------------------------------
__global__ void gather_emb_kernel(const int* __restrict__ sent,
                                  const float* __restrict__ embed,
                                  float* __restrict__ emb_p) {
    int idx = blockIdx.x * blockDim.x + threadIdx.x;
    if (idx < SEQ * EMBP) {
        int t = idx / EMBP, k = idx % EMBP;
        emb_p[idx] = (k < EMB) ? embed[(size_t)sent[t] * EMB + k] : 0.0f;
    }
}

// ---------------------------------------------------------------------------
// K1: transpose recurrent weights -> WhhT[k*1200+j] = Whh[j*300+k]
// (per-step matvec reads coalesced across the 1024 lanes)
// ---------------------------------------------------------------------------
__global__ void transpose_whh_kernel(const float* __restrict__ whh_f,
                                     const float* __restrict__ whh_b,
                                     float* __restrict__ wT_f,
                                     float* __restrict__ wT_b) {
    int idx = blockIdx.x * blockDim.x + threadIdx.x;
    if (idx < 2 * HID * G4) {
        int d = idx / (HID * G4);
        int i = idx % (HID * G4);
        int j = i / HID, k = i % HID;
        const float* src = d ? whh_b : whh_f;
        float*       dst = d ? wT_b  : wT_f;
        dst[k * G4 + j] = src[j * HID + k];
    }
}

// ---------------------------------------------------------------------------
// K2: pack input-projection weights into WMMA-B pair layout:
//   wih2[(k/2)*(G4*2) + j*2 + (k&1)] = W_ih[j*50+k]   (k in [0,52), zero pad)
// -> each lane's B fetch is one aligned global_load_b64, 8B lane stride.
// ---------------------------------------------------------------------------
__global__ void pack_wih_kernel(const float* __restrict__ wih_f,
                                const float* __restrict__ wih_b,
                                float* __restrict__ wih2_f,
                                float* __restrict__ wih2_b) {
    int idx = blockIdx.x * blockDim.x + threadIdx.x;
    if (idx < 2 * EMBP * G4) {
        int d = idx / (EMBP * G4);
        int i = idx % (EMBP * G4);
        int k = i / G4, j = i % G4;
        const float* src = d ? wih_b  : wih_f;
        float*       dst = d ? wih2_b : wih2_f;
        dst[(size_t)(k >> 1) * (G4 * 2) + j * 2 + (k & 1)] =
            (k < EMB) ? src[j * EMB + k] : 0.0f;
    }
}

// ---------------------------------------------------------------------------
// K3: pack tag weights into WMMA-B pair layout (N padded 20->32):
//   wtag2[(k/2)*(TTP*2) + n*2 + (k&1)] = w_tag[n*600+k]  (n>=20 -> 0)
// ---------------------------------------------------------------------------
__global__ void pack_wtag_kernel(const float* __restrict__ wtag,
                                 float* __restrict__ wtag2) {
    int idx = blockIdx.x * blockDim.x + threadIdx.x;
    if (idx < H2 * TTP) {
        int k = idx / TTP, n = idx % TTP;
        wtag2[(size_t)(k >> 1) * (TTP * 2) + n * 2 + (k & 1)] =
            (n < TT) ? wtag[n * H2 + k] : 0.0f;
    }
}

// ---------------------------------------------------------------------------
// K4: input projection via V_WMMA_F32_16X16X4_F32 (exact f32 math).
// xg[t,j] = emb[t'] @ W_ih^T + b_ih + b_hh   (t' reversed for backward dir)
// One wave -> 16(time) x 16(gate) tile; 8 waves/block; fully unrolled,
// one b64 A load + one b64 B load per WMMA, no guards.
// ---------------------------------------------------------------------------
__global__ __launch_bounds__(256)
void proj_wmma_kernel(const float* __restrict__ emb_p,
                      const float* __restrict__ wih2_f, const float* __restrict__ b_ih_f,
                      const float* __restrict__ b_hh_f,
                      const float* __restrict__ wih2_b, const float* __restrict__ b_ih_b,
                      const float* __restrict__ b_hh_b,
                      float* __restrict__ xg_f, float* __restrict__ xg_b) {
    const int dir = blockIdx.z;
    const float* W2 = dir ? wih2_b : wih2_f;
    const float* bi = dir ? b_ih_b : b_ih_f;
    const float* bh = dir ? b_hh_b : b_hh_f;
    float* xg = dir ? xg_b : xg_f;

    const int lane = threadIdx.x & 31;
    const int wave = threadIdx.x >> 5;
    const int t0 = (blockIdx.x * 8 + wave) * 16;
    const int j0 = blockIdx.y * 16;
    const int half = lane >> 4;
    const int l16  = lane & 15;

    const int trow = t0 + l16;
    const int tsrc = dir ? (SEQ - 1 - trow) : trow;
    const int col  = j0 + l16;

    const float* aptr = emb_p + (size_t)tsrc * EMBP;
    const float* bptr = W2 + (size_t)col * 2;

    v8f acc = {};
#pragma unroll
    for (int k = 0; k < EMBP; k += 4) {
        const int ka = k + half * 2;                       // even
        v2f a = *(const v2f*)(aptr + ka);                  // A pair (K=ka,ka+1)
        v2f b = *(const v2f*)(bptr + (size_t)(ka >> 1) * (G4 * 2));
        acc = __builtin_amdgcn_wmma_f32_16x16x4_f32(
            false, a, false, b, (short)0, acc, false, false);
    }
    const float bias = bi[col] + bh[col];
#pragma unroll
    for (int r = 0; r < 8; ++r) {
        int row = t0 + half * 8 + r;                       // C/D: M=r / M=8+r
        xg[(size_t)row * G4 + col] = acc[r] + bias;
    }
}

// ---------------------------------------------------------------------------
// K5: sequential LSTM recurrence. One 1024-thread workgroup per direction
// (grid = 2). Weights stream from L2 (transposed, coalesced); h/c/g in LDS.
// ---------------------------------------------------------------------------
__global__ __launch_bounds__(1024)
void lstm_rec_kernel(const float* __restrict__ xg_f, const float* __restrict__ xg_b,
                     const float* __restrict__ wT_f, const float* __restrict__ wT_b,
                     const float* __restrict__ h0,   const float* __restrict__ c0,
                     float* __restrict__ hsbuf) {
    const int dir = blockIdx.x;
    const float* xg = dir ? xg_b : xg_f;
    const float* Wt = dir ? wT_b : wT_f;

    __shared__ float h_s[HID];
    __shared__ float c_s[HID];
    __shared__ float g_s[G4];

    const int tid = threadIdx.x;
    if (tid < HID) {
        h_s[tid] = h0[dir * HID + tid];
        c_s[tid] = c0[dir * HID + tid];
    }
    __syncthreads();

    for (int t = 0; t < SEQ; ++t) {
        const float* xrow = xg + (size_t)t * G4;
        float a0 = xrow[tid];
        float a1 = (tid < G4 - 1024) ? xrow[tid + 1024] : 0.0f;

        for (int k = 0; k < HID; k += 4) {
            float h0v = h_s[k], h1v = h_s[k + 1], h2v = h_s[k + 2], h3v = h_s[k + 3];
            const float* w = Wt + (size_t)k * G4 + tid;
            a0 = __fmaf_rn(h0v, w[0],      a0);
            a0 = __fmaf_rn(h1v, w[G4],     a0);
            a0 = __fmaf_rn(h2v, w[2 * G4], a0);
            a0 = __fmaf_rn(h3v, w[3 * G4], a0);
            if (tid < G4 - 1024) {
                const float* w2 = w + 1024;
                a1 = __fmaf_rn(h0v, w2[0],      a1);
                a1 = __fmaf_rn(h1v, w2[G4],     a1);
                a1 = __fmaf_rn(h2v, w2[2 * G4], a1);
                a1 = __fmaf_rn(h3v, w2[3 * G4], a1);
            }
        }
        g_s[tid] = a0;
        if (tid < G4 - 1024) g_s[tid + 1024] = a1;
        __syncthreads();

        if (tid < HID) {
            float iv = sigmoidf_(g_s[tid]);
            float fv = sigmoidf_(g_s[tid + HID]);
            float gv = tanhf(g_s[tid + 2 * HID]);
            float ov = sigmoidf_(g_s[tid + 3 * HID]);
            float c  = fv * c_s[tid] + iv * gv;
            float h  = ov * tanhf(c);
            c_s[tid] = c;
            h_s[tid] = h;
            int trow = dir ? (SEQ - 1 - t) : t;        // un-reverse backward dir
            hsbuf[(size_t)trow * H2 + dir * HID + tid] = h;
        }
        __syncthreads();
    }
}

// ---------------------------------------------------------------------------
// K6: tag projection via WMMA: feats[t,n] = lstm_out[t] @ w_tag^T + b_tag
// [16384 x 600] x [600 x 20], N padded to 32, stores guarded to n<20.
// ---------------------------------------------------------------------------
__global__ __launch_bounds__(256)
void tag_proj_wmma_kernel(const float* __restrict__ hsbuf,
                          const float* __restrict__ wtag2,
                          const float* __restrict__ b_tag,
                          float* __restrict__ feats) {
    const int lane = threadIdx.x & 31;
    const int wave = threadIdx.x >> 5;
    const int t0 = (blockIdx.x * 8 + wave) * 16;
    const int n0 = blockIdx.y * 16;
    const int half = lane >> 4;
    const int l16  = lane & 15;

    const int trow = t0 + l16;
    const int col  = n0 + l16;     // 0..31 (padded tag index)

    const float* aptr = hsbuf + (size_t)trow * H2;
    const float* bptr = wtag2 + (size_t)col * 2;

    v8f acc = {};
#pragma unroll 10
    for (int k = 0; k < H2; k += 4) {
        const int ka = k + half * 2;                       // even
        v2f a = *(const v2f*)(aptr + ka);
        v2f b = *(const v2f*)(bptr + (size_t)(ka >> 1) * (TTP * 2));
        acc = __builtin_amdgcn_wmma_f32_16x16x4_f32(
            false, a, false, b, (short)0, acc, false, false);
    }
    if (col < TT) {
        const float bias = b_tag[col];
#pragma unroll
        for (int r = 0; r < 8; ++r) {
            int row = t0 + half * 8 + r;
            feats[(size_t)row * TT + col] = acc[r] + bias;
        }
    }
}

// ---------------------------------------------------------------------------
// K7: Viterbi scan + termination + backtrack. Single wave32 (20 lanes active);
// transitions row per lane in registers (fully unrolled, no scratch).
// ---------------------------------------------------------------------------
__global__ __launch_bounds__(32)
void viterbi_kernel(const float* __restrict__ feats,
                    const float* __restrict__ trans,
                    signed char* __restrict__ bps,
                    float* __restrict__ out) {
    __shared__ float fv[TT];
    __shared__ float term[TT];
    const int lane = threadIdx.x;

    float tr[TT];
    if (lane < TT) {
#pragma unroll
        for (int p = 0; p < TT; ++p) tr[p] = trans[lane * TT + p];
        fv[lane] = (lane == START_TAG) ? 0.0f : NEGV;
    }
    __syncthreads();

    for (int t = 0; t < SEQ; ++t) {
        float best = -3.4e38f;
        int arg = 0;
        if (lane < TT) {
#pragma unroll
            for (int p = 0; p < TT; ++p) {
                float v = fv[p] + tr[p];
                if (v > best) { best = v; arg = p; }   // first-max like jnp.argmax
            }
        }
        __syncthreads();
        if (lane < TT) {
            fv[lane] = best + feats[t * TT + lane];
            bps[t * TT + lane] = (signed char)arg;
        }
        __syncthreads();
    }

    if (lane < TT) term[lane] = fv[lane] + trans[STOP_TAG * TT + lane];
    __syncthreads();

    if (lane == 0) {
        int best = 0;
        float score = term[0];
        for (int n = 1; n < TT; ++n)
            if (term[n] > score) { score = term[n]; best = n; }
        out[0] = score;
        int cur = best;
        out[1 + (SEQ - 1)] = (float)cur;            // path[SEQ-1] = best
        for (int t = SEQ - 1; t >= 1; --t) {        // path[t-1] = bps[t][path[t]]
            cur = (int)bps[t * TT + cur];
            out[t] = (float)cur;
        }
    }
}

// ---------------------------------------------------------------------------
extern "C" void kernel_launch(void* const* d_in, const int* in_sizes, int n_in,
                              void* d_out, int out_size, void* d_ws, size_t ws_size,
                              hipStream_t stream) {
    (void)in_sizes; (void)n_in; (void)out_size; (void)ws_size;

    const int*   sent  = (const int*)  d_in[0];
    const float* embed = (const float*)d_in[1];
    const float* wihf  = (const float*)d_in[2];
    const float* whhf  = (const float*)d_in[3];
    const float* bihf  = (const float*)d_in[4];
    const float* bhhf  = (const float*)d_in[5];
    const float* wihb  = (const float*)d_in[6];
    const float* whhb  = (const float*)d_in[7];
    const float* bihb  = (const float*)d_in[8];
    const float* bhhb  = (const float*)d_in[9];
    const float* h0    = (const float*)d_in[10];
    const float* c0    = (const float*)d_in[11];
    const float* wtag  = (const float*)d_in[12];
    const float* btag  = (const float*)d_in[13];
    const float* trans = (const float*)d_in[14];

    // workspace carve-up (floats)
    float* ws     = (float*)d_ws;
    float* emb_p  = ws;                                  // SEQ*EMBP
    float* xg_f   = emb_p  + (size_t)SEQ * EMBP;         // SEQ*G4
    float* xg_b   = xg_f   + (size_t)SEQ * G4;           // SEQ*G4
    float* wT_f   = xg_b   + (size_t)SEQ * G4;           // HID*G4
    float* wT_b   = wT_f   + (size_t)HID * G4;           // HID*G4
    float* wih2_f = wT_b   + (size_t)HID * G4;           // EMBP*G4
    float* wih2_b = wih2_f + (size_t)EMBP * G4;          // EMBP*G4
    float* wtag2  = wih2_b + (size_t)EMBP * G4;          // H2*TTP
    float* hsbuf  = wtag2  + (size_t)H2 * TTP;           // SEQ*H2
    float* feats  = hsbuf  + (size_t)SEQ * H2;           // SEQ*TT
    signed char* bps = (signed char*)(feats + (size_t)SEQ * TT);   // SEQ*TT bytes

    gather_emb_kernel<<<(SEQ * EMBP + 255) / 256, 256, 0, stream>>>(sent, embed, emb_p);

    transpose_whh_kernel<<<(2 * HID * G4 + 255) / 256, 256, 0, stream>>>(
        whhf, whhb, wT_f, wT_b);

    pack_wih_kernel<<<(2 * EMBP * G4 + 255) / 256, 256, 0, stream>>>(
        wihf, wihb, wih2_f, wih2_b);

    pack_wtag_kernel<<<(H2 * TTP + 255) / 256, 256, 0, stream>>>(wtag, wtag2);

    dim3 gp(SEQ / (8 * 16), G4 / 16, 2);                 // 128 x 75 x 2
    proj_wmma_kernel<<<gp, 256, 0, stream>>>(
        emb_p, wih2_f, bihf, bhhf, wih2_b, bihb, bhhb, xg_f, xg_b);

    lstm_rec_kernel<<<2, 1024, 0, stream>>>(xg_f, xg_b, wT_f, wT_b, h0, c0, hsbuf);

    dim3 gt(SEQ / (8 * 16), TTP / 16);                   // 128 x 2
    tag_proj_wmma_kernel<<<gt, 256, 0, stream>>>(hsbuf, wtag2, btag, feats);

    viterbi_kernel<<<1, 32, 0, stream>>>(feats, trans, bps, (float*)d_out);
}